// GAT_8916352106937
// MI455X (gfx1250) — compile-verified
//
#include <hip/hip_runtime.h>
#include <hip/hip_bf16.h>
#include <math.h>

typedef float v2f __attribute__((ext_vector_type(2)));
typedef float v8f __attribute__((ext_vector_type(8)));

#define NN 100000
#define EE_EDGES 1600000
#define ETOT (EE_EDGES + NN)
#define IND 128
#define H1 8
#define F1 16
#define NEG_SLOPE 0.2f

// ---------------------------------------------------------------------------
// GEMM1: h1[N,128] = x[N,128] @ W[128,128], fp32 WMMA 16x16x4.
// One block = one 16-row tile (staged in LDS), 8 waves = 8 column tiles.
// ---------------------------------------------------------------------------
__global__ void gat_gemm1_wmma(const float* __restrict__ x,
                               const float* __restrict__ W,
                               float* __restrict__ h1) {
    __shared__ float xs[16][132];  // 132 stride -> conflict-free column reads
    const int m0 = blockIdx.x * 16;
    const int tid = threadIdx.x;

    // cooperative load: 16x128 floats, 8 per thread
    {
        int idx = tid * 8;
        int r = idx >> 7;
        int c = idx & 127;
        const float4* src = (const float4*)(x + (size_t)(m0 + r) * IND + c);
        float4 a = src[0];
        float4 b = src[1];
        xs[r][c + 0] = a.x; xs[r][c + 1] = a.y; xs[r][c + 2] = a.z; xs[r][c + 3] = a.w;
        xs[r][c + 4] = b.x; xs[r][c + 5] = b.y; xs[r][c + 6] = b.z; xs[r][c + 7] = b.w;
    }
    __syncthreads();

    const int wave = tid >> 5;      // 0..7 -> column tile (== head)
    const int lane = tid & 31;
    const int half = lane >> 4;
    const int lr   = lane & 15;
    const int col0 = wave * 16;

    v8f c = {};
    for (int k = 0; k < IND; k += 4) {
        const int kb = k + half * 2;
        v2f a, b;
        a.x = xs[lr][kb];
        a.y = xs[lr][kb + 1];
        b.x = W[(size_t)kb * 128 + col0 + lr];
        b.y = W[(size_t)(kb + 1) * 128 + col0 + lr];
        c = __builtin_amdgcn_wmma_f32_16x16x4_f32(false, a, false, b,
                                                  (short)0, c, false, false);
    }
#pragma unroll
    for (int r = 0; r < 8; ++r) {
        h1[(size_t)(m0 + r + half * 8) * 128 + col0 + lr] = c[r];
    }
}

// ---------------------------------------------------------------------------
// GEMM2: h2[N,16] = x2[N,128] @ W2[128,16], fp32 WMMA. 8 waves = 8 row tiles.
// ---------------------------------------------------------------------------
__global__ void gat_gemm2_wmma(const float* __restrict__ x2,
                               const float* __restrict__ W2,
                               float* __restrict__ h2, int n) {
    const int tid = threadIdx.x;
    const int wave = tid >> 5;
    const int lane = tid & 31;
    const int half = lane >> 4;
    const int lr   = lane & 15;
    const int m0 = blockIdx.x * 128 + wave * 16;
    const int row = m0 + lr;
    const bool rok = row < n;

    v8f c = {};
    for (int k = 0; k < IND; k += 4) {
        const int kb = k + half * 2;
        v2f a, b;
        a.x = rok ? x2[(size_t)row * 128 + kb]     : 0.0f;
        a.y = rok ? x2[(size_t)row * 128 + kb + 1] : 0.0f;
        b.x = W2[kb * 16 + lr];
        b.y = W2[(kb + 1) * 16 + lr];
        c = __builtin_amdgcn_wmma_f32_16x16x4_f32(false, a, false, b,
                                                  (short)0, c, false, false);
    }
#pragma unroll
    for (int r = 0; r < 8; ++r) {
        const int row2 = m0 + r + half * 8;
        if (row2 < n) h2[(size_t)row2 * 16 + lr] = c[r];
    }
}

// ---------------------------------------------------------------------------
// alpha_s[n,h] = sum_f h[n,h,f]*att_src[h,f] ; same for alpha_d. F = 16.
// ---------------------------------------------------------------------------
__global__ void gat_alpha(const float* __restrict__ hfeat,
                          const float* __restrict__ a_src,
                          const float* __restrict__ a_dst,
                          float* __restrict__ as_out,
                          float* __restrict__ ad_out,
                          int H, int total) {
    int tid = blockIdx.x * blockDim.x + threadIdx.x;
    if (tid >= total) return;
    int nn = tid / H;
    int h  = tid - nn * H;
    const float4* hp = (const float4*)(hfeat + ((size_t)nn * H + h) * 16);
    const float4* sp = (const float4*)(a_src + h * 16);
    const float4* dp = (const float4*)(a_dst + h * 16);
    float s = 0.f, d = 0.f;
#pragma unroll
    for (int q = 0; q < 4; ++q) {
        float4 hv = hp[q], sv = sp[q], dv = dp[q];
        s += hv.x * sv.x + hv.y * sv.y + hv.z * sv.z + hv.w * sv.w;
        d += hv.x * dv.x + hv.y * dv.y + hv.z * dv.z + hv.w * dv.w;
    }
    as_out[tid] = s;
    ad_out[tid] = d;
}

// ---------------------------------------------------------------------------
// Pass A: e = leaky_relu(as[src]+ad[dst]); ee = exp(e); denom[dst] += ee.
// (segment-max skipped: softmax is shift-invariant and logits are O(1))
// ---------------------------------------------------------------------------
__global__ void gat_edge_softmax(const float* __restrict__ as_,
                                 const float* __restrict__ ad_,
                                 const int* __restrict__ srcs,
                                 const int* __restrict__ dsts,
                                 float* __restrict__ ee,
                                 float* __restrict__ den,
                                 int nE, int nEtot, int H) {
    int tid = blockIdx.x * blockDim.x + threadIdx.x;
    if (tid >= nEtot * H) return;
    int e = tid / H;
    int h = tid - e * H;
    int s, d;
    if (e < nE) { s = srcs[e]; d = dsts[e]; }
    else        { s = d = e - nE; }
    float v = as_[(size_t)s * H + h] + ad_[(size_t)d * H + h];
    v = v > 0.f ? v : NEG_SLOPE * v;
    float ex = expf(v);
    ee[tid] = ex;
    atomicAdd(den + (size_t)d * H + h, ex);
}

// ---------------------------------------------------------------------------
// Pass B: out[dst,h,:] += (ee/denom[dst,h]) * hfeat[src,h,:]
// 16 lanes per (edge,head): contiguous 64B gather + contiguous 64B atomics.
// ---------------------------------------------------------------------------
__global__ void gat_edge_message(const float* __restrict__ hfeat,
                                 const float* __restrict__ ee,
                                 const float* __restrict__ den,
                                 const int* __restrict__ srcs,
                                 const int* __restrict__ dsts,
                                 float* __restrict__ outa,
                                 int nE, int nEtot, int H) {
    int tid = blockIdx.x * blockDim.x + threadIdx.x;
    int g = tid >> 4;
    int f = tid & 15;
    if (g >= nEtot * H) return;
    int e = g / H;
    int h = g - e * H;
    int s, d;
    if (e < nE) { s = srcs[e]; d = dsts[e]; }
    else        { s = d = e - nE; }
    float w = ee[g] / den[(size_t)d * H + h];
    float val = hfeat[((size_t)s * H + h) * 16 + f] * w;
    atomicAdd(outa + ((size_t)d * H + h) * 16 + f, val);
}

// ---------------------------------------------------------------------------
// x2 = elu(out1 + b1), in place
// ---------------------------------------------------------------------------
__global__ void gat_bias_elu(float* __restrict__ xio,
                             const float* __restrict__ b, int total) {
    int i = blockIdx.x * blockDim.x + threadIdx.x;
    if (i >= total) return;
    float v = xio[i] + b[i & 127];
    xio[i] = v > 0.f ? v : (expf(v) - 1.0f);
}

// ---------------------------------------------------------------------------
// d_out[n,:] = log_softmax(out2[n,:] + b2)
// ---------------------------------------------------------------------------
__global__ void gat_logsoftmax(const float* __restrict__ out2,
                               const float* __restrict__ b2,
                               float* __restrict__ dout, int n) {
    int nn = blockIdx.x * blockDim.x + threadIdx.x;
    if (nn >= n) return;
    float v[16];
    const float4* p = (const float4*)(out2 + (size_t)nn * 16);
    const float4* bp = (const float4*)b2;
#pragma unroll
    for (int q = 0; q < 4; ++q) {
        float4 a = p[q], bb = bp[q];
        v[q * 4 + 0] = a.x + bb.x; v[q * 4 + 1] = a.y + bb.y;
        v[q * 4 + 2] = a.z + bb.z; v[q * 4 + 3] = a.w + bb.w;
    }
    float m = v[0];
#pragma unroll
    for (int f = 1; f < 16; ++f) m = fmaxf(m, v[f]);
    float sum = 0.f;
#pragma unroll
    for (int f = 0; f < 16; ++f) sum += expf(v[f] - m);
    float lse = m + logf(sum);
#pragma unroll
    for (int f = 0; f < 16; ++f) dout[(size_t)nn * 16 + f] = v[f] - lse;
}

// ---------------------------------------------------------------------------
extern "C" void kernel_launch(void* const* d_in, const int* in_sizes, int n_in,
                              void* d_out, int out_size, void* d_ws, size_t ws_size,
                              hipStream_t stream) {
    const float* x      = (const float*)d_in[0];
    const int*   ei     = (const int*)d_in[1];   // [2, E] (jax default int32)
    const float* W1     = (const float*)d_in[2];
    const float* atts1  = (const float*)d_in[3];
    const float* attd1  = (const float*)d_in[4];
    const float* b1     = (const float*)d_in[5];
    const float* W2     = (const float*)d_in[6];
    const float* atts2  = (const float*)d_in[7];
    const float* attd2  = (const float*)d_in[8];
    const float* b2     = (const float*)d_in[9];
    float* outp = (float*)d_out;
    float* ws   = (float*)d_ws;

    const int* srcs = ei;
    const int* dsts = ei + EE_EDGES;

    // workspace layout (floats)
    const size_t oH1   = 0;                       // N*128 (h1; layer2: h2 @0, out2 @N*16)
    const size_t oX2   = (size_t)NN * 128;        // N*128 (out1 accum -> x2)
    const size_t oAS1  = oX2 + (size_t)NN * 128;  // N*8
    const size_t oAD1  = oAS1 + (size_t)NN * 8;   // N*8
    const size_t oDEN1 = oAD1 + (size_t)NN * 8;   // N*8
    const size_t oEE   = oDEN1 + (size_t)NN * 8;  // Etot*8 (reused as ee2)

    float* h1   = ws + oH1;
    float* x2   = ws + oX2;
    float* as1  = ws + oAS1;
    float* ad1  = ws + oAD1;
    float* den1 = ws + oDEN1;
    float* eeb  = ws + oEE;
    // layer-2 aliases (h1/as1/ad1/den1 dead after layer-1 message pass)
    float* h2   = ws + oH1;
    float* out2 = ws + oH1 + (size_t)NN * 16;
    float* as2  = ws + oAS1;
    float* ad2  = ws + oAD1;
    float* den2 = ws + oDEN1;

    const int B = 256;

    // ---- layer 1 ----
    hipMemsetAsync(x2,   0, (size_t)NN * 128 * sizeof(float), stream); // out1 accum
    hipMemsetAsync(den1, 0, (size_t)NN * 8 * sizeof(float), stream);

    gat_gemm1_wmma<<<NN / 16, B, 0, stream>>>(x, W1, h1);
    gat_alpha<<<(NN * H1 + B - 1) / B, B, 0, stream>>>(h1, atts1, attd1, as1, ad1, H1, NN * H1);
    gat_edge_softmax<<<(ETOT * H1 + B - 1) / B, B, 0, stream>>>(
        as1, ad1, srcs, dsts, eeb, den1, EE_EDGES, ETOT, H1);
    gat_edge_message<<<((size_t)ETOT * H1 * 16 + B - 1) / B, B, 0, stream>>>(
        h1, eeb, den1, srcs, dsts, x2, EE_EDGES, ETOT, H1);
    gat_bias_elu<<<(NN * 128 + B - 1) / B, B, 0, stream>>>(x2, b1, NN * 128);

    // ---- layer 2 ----
    hipMemsetAsync(out2, 0, (size_t)NN * 16 * sizeof(float), stream);
    hipMemsetAsync(den2, 0, (size_t)NN * sizeof(float), stream);

    gat_gemm2_wmma<<<(NN + 127) / 128, B, 0, stream>>>(x2, W2, h2, NN);
    gat_alpha<<<(NN + B - 1) / B, B, 0, stream>>>(h2, atts2, attd2, as2, ad2, 1, NN);
    gat_edge_softmax<<<(ETOT + B - 1) / B, B, 0, stream>>>(
        as2, ad2, srcs, dsts, eeb, den2, EE_EDGES, ETOT, 1);
    gat_edge_message<<<((size_t)ETOT * 16 + B - 1) / B, B, 0, stream>>>(
        h2, eeb, den2, srcs, dsts, out2, EE_EDGES, ETOT, 1);
    gat_logsoftmax<<<(NN + B - 1) / B, B, 0, stream>>>(out2, b2, outp, NN);
}